// StructuralProbe_2911987827409
// MI455X (gfx1250) — compile-verified
//
#include <hip/hip_runtime.h>

typedef __attribute__((ext_vector_type(2))) float v2f;
typedef __attribute__((ext_vector_type(8))) float v8f;

#define MODEL_DIM 1024
#define RANK      128
#define SEQ       512
#define BATCH     8

// ---------------------------------------------------------------------------
// Kernel 1: T = batch @ proj.  M = BATCH*SEQ = 4096 rows, K = 1024, N = 128.
// One wave (32 lanes) per 16x16 output tile, V_WMMA_F32_16X16X4_F32, K step 4.
//
// A 16x4 f32 layout: lane l -> row l%16, K = {2*(l/16), 2*(l/16)+1}  (v2f)
// B 4x16 f32 layout: lane l -> col l%16, K = {2*(l/16), 2*(l/16)+1}  (v2f)
// D 16x16 f32 layout: vgpr r, lane l -> (M = r + 8*(l/16), N = l%16)
// ---------------------------------------------------------------------------
__global__ __launch_bounds__(32) void proj_wmma_f32(
    const float* __restrict__ A,   // (4096, 1024) row-major
    const float* __restrict__ P,   // (1024, 128)  row-major
    float* __restrict__ T)         // (4096, 128)  row-major
{
  const int lane = threadIdx.x;
  const int half = lane >> 4;          // 0 or 1  -> K sub-pair
  const int l16  = lane & 15;          // M (for A) / N (for B)
  const int i0   = blockIdx.x * 16;    // M tile base
  const int n0   = blockIdx.y * 16;    // N tile base

  const float* arow = A + (size_t)(i0 + l16) * MODEL_DIM + 2 * half;
  const float* bcol = P + (size_t)(2 * half) * RANK + (n0 + l16);

  v8f c = {};
  for (int k = 0; k < MODEL_DIM; k += 4) {
    // A: two consecutive f32 of this lane's row -> global_load_b64
    v2f a = *(const v2f*)(arow + k);
    // B: proj[k + 2*half + {0,1}][n0 + l16] (stride RANK between the pair)
    v2f b;
    b.x = bcol[(size_t)k * RANK];
    b.y = bcol[(size_t)(k + 1) * RANK];
    c = __builtin_amdgcn_wmma_f32_16x16x4_f32(
            /*neg_a=*/false, a, /*neg_b=*/false, b,
            /*c_mod=*/(short)0, c, /*reuse_a=*/false, /*reuse_b=*/false);
  }

  float* trow = T + (size_t)i0 * RANK + n0;
#pragma unroll
  for (int r = 0; r < 8; ++r) {
    const int m = r + 8 * half;
    trow[(size_t)m * RANK + l16] = c[r];
  }
}

// ---------------------------------------------------------------------------
// Kernel 2: rn[i] = sum_k T[i][k]^2.  One thread per row (4096 rows of 128).
// ---------------------------------------------------------------------------
__global__ void rownorm_kernel(const float* __restrict__ T,
                               float* __restrict__ rn)
{
  const int i = blockIdx.x * blockDim.x + threadIdx.x;
  if (i < BATCH * SEQ) {
    const float* t = T + (size_t)i * RANK;
    float s = 0.0f;
#pragma unroll 4
    for (int k = 0; k < RANK; ++k) {
      const float v = t[k];
      s += v * v;
    }
    rn[i] = s;
  }
}

// ---------------------------------------------------------------------------
// Kernel 3: out[b,i,j] = rn[i] + rn[j] - 2 * (T_b T_b^T)[i,j], clamped >= 0.
// One wave per 16x16 (i,j) tile; Gram via WMMA f32, K = RANK = 128.
// B = T^T, so B's column n is T's row (j0+n): A and B share the same
// per-lane contiguous b64 load pattern from T.
// ---------------------------------------------------------------------------
__global__ __launch_bounds__(32) void dist_wmma_f32(
    const float* __restrict__ T,   // (BATCH*SEQ, RANK)
    const float* __restrict__ rn,  // (BATCH*SEQ)
    float* __restrict__ out)       // (BATCH, SEQ, SEQ)
{
  const int lane = threadIdx.x;
  const int half = lane >> 4;
  const int l16  = lane & 15;
  const int b    = blockIdx.z;
  const int j0   = blockIdx.x * 16;
  const int i0   = blockIdx.y * 16;

  const float* Tb   = T + (size_t)b * SEQ * RANK;
  const float* arow = Tb + (size_t)(i0 + l16) * RANK + 2 * half;
  const float* brow = Tb + (size_t)(j0 + l16) * RANK + 2 * half;

  v8f c = {};
#pragma unroll 4
  for (int k = 0; k < RANK; k += 4) {
    v2f a  = *(const v2f*)(arow + k);
    v2f bb = *(const v2f*)(brow + k);
    c = __builtin_amdgcn_wmma_f32_16x16x4_f32(
            false, a, false, bb, (short)0, c, false, false);
  }

  const float* rnb = rn + (size_t)b * SEQ;
  const float  rj  = rnb[j0 + l16];
  float* outb = out + ((size_t)b * SEQ + (size_t)i0) * SEQ + j0;
#pragma unroll
  for (int r = 0; r < 8; ++r) {
    const int m = r + 8 * half;
    const float d = rnb[i0 + m] + rj - 2.0f * c[r];
    outb[(size_t)m * SEQ + l16] = fmaxf(d, 0.0f);
  }
}

// ---------------------------------------------------------------------------
extern "C" void kernel_launch(void* const* d_in, const int* in_sizes, int n_in,
                              void* d_out, int out_size, void* d_ws, size_t ws_size,
                              hipStream_t stream)
{
  const float* batch = (const float*)d_in[0];  // (8, 512, 1024) f32
  const float* proj  = (const float*)d_in[1];  // (1024, 128)    f32
  float* out = (float*)d_out;                  // (8, 512, 512)  f32

  float* T  = (float*)d_ws;                    // 4096*128 f32 = 2 MB
  float* rn = T + (size_t)BATCH * SEQ * RANK;  // 4096 f32

  // Projection GEMM: 256 M-tiles x 8 N-tiles, one wave each.
  dim3 g1(BATCH * SEQ / 16, RANK / 16);
  proj_wmma_f32<<<g1, 32, 0, stream>>>(batch, proj, T);

  // Row norms.
  rownorm_kernel<<<(BATCH * SEQ + 255) / 256, 256, 0, stream>>>(T, rn);

  // Distance tiles: 32x32 tiles per batch, 8 batches.
  dim3 g2(SEQ / 16, SEQ / 16, BATCH);
  dist_wmma_f32<<<g2, 32, 0, stream>>>(T, rn, out);
}